// GraphVAE_3358664425543
// MI455X (gfx1250) — compile-verified
//
#include <hip/hip_runtime.h>
#include <hip/hip_bf16.h>
#include <math.h>

typedef __attribute__((ext_vector_type(2))) float v2f;
typedef __attribute__((ext_vector_type(8))) float v8f;

#define N_NODES   100000
#define N_EDGES   1600000
#define N_GRAPHS  200
#define NPG       500
#define IN_DIM    24
#define HID       64
#define LAT       32
#define DEC_H     128

// ---------------- degree / normalization ----------------
__global__ void k_fill1(float* __restrict__ p, int n) {
    int i = blockIdx.x * blockDim.x + threadIdx.x;
    if (i < n) p[i] = 1.0f;   // self-loop contributes 1 to degree
}

__global__ void k_deg_scatter(const int* __restrict__ col, float* __restrict__ deg, int e) {
    int i = blockIdx.x * blockDim.x + threadIdx.x;
    if (i < e) atomicAdd(&deg[col[i]], 1.0f);
}

__global__ void k_rsqrt_inplace(float* __restrict__ p, int n) {
    int i = blockIdx.x * blockDim.x + threadIdx.x;
    if (i < n) p[i] = rsqrtf(p[i]);   // deg >= 1 always
}

// ---------------- dense GEMM via f32 WMMA (16x16x4), LDS-staged ----------------
// OUT[M,64] = X[M,K] @ W[K,64].  One block = 128 threads = 4 waves.
// Block b owns rows m0=16b; wave w owns the 16-col tile n0=16w.
// W (K*64) and the 16xK A-tile are staged in LDS once per block, then each
// wave feeds fragments to v_wmma_f32_16x16x4_f32 via ds loads. K is a
// compile-time constant so the K loop fully unrolls (6 or 16 WMMAs).
template <int K>
__global__ void k_gemm_wmma_f32(const float* __restrict__ X, const float* __restrict__ W,
                                float* __restrict__ OUT) {
    __shared__ float sW[K * HID];   // up to 16 KB
    __shared__ float sA[16 * K];    // up to 4 KB

    const int t  = threadIdx.x;     // 0..127
    const int m0 = blockIdx.x * 16;

    // cooperative coalesced staging (float4 granularity)
    const float4* gW = (const float4*)W;
    const float4* gA = (const float4*)(X + (size_t)m0 * K);
    for (int idx = t; idx < (K * HID) / 4; idx += 128) ((float4*)sW)[idx] = gW[idx];
    for (int idx = t; idx < (16 * K) / 4; idx += 128) ((float4*)sA)[idx] = gA[idx];
    __syncthreads();

    const int wave = t >> 5;
    const int n0   = wave * 16;
    const int lane = t & 31;
    const int half = lane >> 4;     // 0: lanes 0-15, 1: lanes 16-31
    const int l16  = lane & 15;

    v8f acc = {};
#pragma unroll
    for (int k0 = 0; k0 < K; k0 += 4) {
        const int ka = k0 + (half << 1);     // upper lanes hold K+2,K+3
        v2f a, b;
        a.x = sA[l16 * K + ka];
        a.y = sA[l16 * K + ka + 1];
        b.x = sW[(ka)     * HID + n0 + l16];
        b.y = sW[(ka + 1) * HID + n0 + l16];
        acc = __builtin_amdgcn_wmma_f32_16x16x4_f32(false, a, false, b,
                                                    (short)0, acc, false, false);
    }
#pragma unroll
    for (int v = 0; v < 8; ++v)
        OUT[(size_t)(m0 + v + half * 8) * HID + (n0 + l16)] = acc[v];
}

// ---------------- GCN aggregation ----------------
// init: OUT[i,f] = H[i,f] * dinv[i]^2   (self-loop term)
__global__ void k_agg_init(const float* __restrict__ H, const float* __restrict__ dinv,
                           float* __restrict__ OUT) {
    int i = blockIdx.x * blockDim.x + threadIdx.x;   // N_NODES*HID threads
    if (i >= N_NODES * HID) return;
    int node = i >> 6;
    float di = dinv[node];
    OUT[i] = H[i] * di * di;
}

// edge scatter: 16 threads per edge, float4 gather + 4 f32 atomics each
__global__ void k_edge_scatter(const int* __restrict__ rows, const int* __restrict__ cols,
                               const float* __restrict__ dinv, const float* __restrict__ H,
                               float* __restrict__ OUT) {
    long gid = (long)blockIdx.x * blockDim.x + threadIdx.x;
    int e = (int)(gid >> 4);
    if (e >= N_EDGES) return;
    int fq = ((int)gid & 15) * 4;
    int r = rows[e], c = cols[e];
    float norm = dinv[r] * dinv[c];
    const float4 hv = *reinterpret_cast<const float4*>(H + (size_t)r * HID + fq);
    float* dst = OUT + (size_t)c * HID + fq;
    atomicAdd(dst + 0, hv.x * norm);
    atomicAdd(dst + 1, hv.y * norm);
    atomicAdd(dst + 2, hv.z * norm);
    atomicAdd(dst + 3, hv.w * norm);
}

__global__ void k_bias_relu(float* __restrict__ H, const float* __restrict__ b) {
    int i = blockIdx.x * blockDim.x + threadIdx.x;
    if (i >= N_NODES * HID) return;
    float v = H[i] + b[i & (HID - 1)];
    H[i] = v > 0.0f ? v : 0.0f;
}

// ---------------- pooling / latent / decoder ----------------
__global__ void k_pool_mean(const float* __restrict__ H, float* __restrict__ pooled) {
    int g = blockIdx.x;          // 200 blocks
    int f = threadIdx.x;         // 64 threads
    const float* base = H + (size_t)g * NPG * HID;
    float s = 0.0f;
    for (int n = 0; n < NPG; ++n) s += base[(size_t)n * HID + f];
    pooled[g * HID + f] = s * (1.0f / (float)NPG);
}

__global__ void k_latent(const float* __restrict__ pooled,
                         const float* __restrict__ Wmu, const float* __restrict__ bmu,
                         const float* __restrict__ Wlv, const float* __restrict__ blv,
                         const float* __restrict__ eps,
                         float* __restrict__ mu_out, float* __restrict__ lv_out,
                         float* __restrict__ z) {
    int i = blockIdx.x * blockDim.x + threadIdx.x;   // 200*32
    if (i >= N_GRAPHS * LAT) return;
    int g = i >> 5, j = i & (LAT - 1);
    const float* p = pooled + g * HID;
    float mu = bmu[j], lv = blv[j];
    for (int k = 0; k < HID; ++k) {
        float pk = p[k];
        mu += pk * Wmu[k * LAT + j];
        lv += pk * Wlv[k * LAT + j];
    }
    mu_out[i] = mu;
    lv_out[i] = lv;
    z[i] = mu + eps[i] * __expf(0.5f * lv);
}

// decoder operates on the 200 unique z rows only (repeat happens at the end)
__global__ void k_dec1(const float* __restrict__ z, const float* __restrict__ Wd1,
                       const float* __restrict__ bd1, float* __restrict__ hd) {
    int i = blockIdx.x * blockDim.x + threadIdx.x;   // 200*128
    if (i >= N_GRAPHS * DEC_H) return;
    int g = i >> 7, h = i & (DEC_H - 1);
    const float* zr = z + g * LAT;
    float s = bd1[h];
    for (int k = 0; k < LAT; ++k) s += zr[k] * Wd1[k * DEC_H + h];
    hd[i] = s > 0.0f ? s : 0.0f;
}

__global__ void k_dec2(const float* __restrict__ hd, const float* __restrict__ Wd2,
                       const float* __restrict__ bd2, float* __restrict__ rs) {
    int i = blockIdx.x * blockDim.x + threadIdx.x;   // 200*24
    if (i >= N_GRAPHS * IN_DIM) return;
    int g = i / IN_DIM, o = i % IN_DIM;
    const float* hr = hd + g * DEC_H;
    float s = bd2[o];
    for (int k = 0; k < DEC_H; ++k) s += hr[k] * Wd2[k * IN_DIM + o];
    rs[i] = 1.0f / (1.0f + __expf(-s));
}

__global__ void k_broadcast(const float* __restrict__ rs, float* __restrict__ out) {
    int i = blockIdx.x * blockDim.x + threadIdx.x;   // 100000*24
    if (i >= N_NODES * IN_DIM) return;
    int node = i / IN_DIM, o = i % IN_DIM;
    out[i] = rs[(node / NPG) * IN_DIM + o];
}

// ---------------- launcher ----------------
extern "C" void kernel_launch(void* const* d_in, const int* in_sizes, int n_in,
                              void* d_out, int out_size, void* d_ws, size_t ws_size,
                              hipStream_t stream) {
    const float* x   = (const float*)d_in[0];
    const int*   ei  = (const int*)d_in[1];
    const float* eps = (const float*)d_in[3];
    const float* W1  = (const float*)d_in[4];
    const float* b1  = (const float*)d_in[5];
    const float* W2  = (const float*)d_in[6];
    const float* b2  = (const float*)d_in[7];
    const float* Wmu = (const float*)d_in[8];
    const float* bmu = (const float*)d_in[9];
    const float* Wlv = (const float*)d_in[10];
    const float* blv = (const float*)d_in[11];
    const float* Wd1 = (const float*)d_in[12];
    const float* bd1 = (const float*)d_in[13];
    const float* Wd2 = (const float*)d_in[14];
    const float* bd2 = (const float*)d_in[15];
    float* out = (float*)d_out;

    const int* rows = ei;            // edge_index[0]
    const int* cols = ei + N_EDGES;  // edge_index[1]

    float* ws     = (float*)d_ws;
    float* bufA   = ws;                                  // [N_NODES, HID]
    float* bufB   = bufA + (size_t)N_NODES * HID;        // [N_NODES, HID]
    float* dinv   = bufB + (size_t)N_NODES * HID;        // [N_NODES]
    float* pooled = dinv + N_NODES;                      // [200, 64]
    float* zb     = pooled + N_GRAPHS * HID;             // [200, 32]
    float* hd     = zb + N_GRAPHS * LAT;                 // [200, 128]
    float* rs     = hd + N_GRAPHS * DEC_H;               // [200, 24]

    float* mu_out = out + (size_t)N_NODES * IN_DIM;
    float* lv_out = mu_out + N_GRAPHS * LAT;

    const int NF = N_NODES * HID;                        // 6.4M
    const int gemm_blocks = N_NODES / 16;                // 6250, 4 waves/block

    // degrees -> dinv
    k_fill1<<<(N_NODES + 255) / 256, 256, 0, stream>>>(dinv, N_NODES);
    k_deg_scatter<<<(N_EDGES + 255) / 256, 256, 0, stream>>>(cols, dinv, N_EDGES);
    k_rsqrt_inplace<<<(N_NODES + 255) / 256, 256, 0, stream>>>(dinv, N_NODES);

    // conv1: bufA = x@W1 ; bufB = aggregate(bufA) ; +b1, relu
    k_gemm_wmma_f32<IN_DIM><<<gemm_blocks, 128, 0, stream>>>(x, W1, bufA);
    k_agg_init<<<(NF + 255) / 256, 256, 0, stream>>>(bufA, dinv, bufB);
    k_edge_scatter<<<(N_EDGES * 16) / 256, 256, 0, stream>>>(rows, cols, dinv, bufA, bufB);
    k_bias_relu<<<(NF + 255) / 256, 256, 0, stream>>>(bufB, b1);

    // conv2: bufA = bufB@W2 ; bufB = aggregate(bufA) ; +b2, relu
    k_gemm_wmma_f32<HID><<<gemm_blocks, 128, 0, stream>>>(bufB, W2, bufA);
    k_agg_init<<<(NF + 255) / 256, 256, 0, stream>>>(bufA, dinv, bufB);
    k_edge_scatter<<<(N_EDGES * 16) / 256, 256, 0, stream>>>(rows, cols, dinv, bufA, bufB);
    k_bias_relu<<<(NF + 255) / 256, 256, 0, stream>>>(bufB, b2);

    // pool -> latent -> decoder(200 rows) -> broadcast
    k_pool_mean<<<N_GRAPHS, HID, 0, stream>>>(bufB, pooled);
    k_latent<<<(N_GRAPHS * LAT + 255) / 256, 256, 0, stream>>>(
        pooled, Wmu, bmu, Wlv, blv, eps, mu_out, lv_out, zb);
    k_dec1<<<(N_GRAPHS * DEC_H + 255) / 256, 256, 0, stream>>>(zb, Wd1, bd1, hd);
    k_dec2<<<(N_GRAPHS * IN_DIM + 255) / 256, 256, 0, stream>>>(hd, Wd2, bd2, rs);
    k_broadcast<<<(N_NODES * IN_DIM + 255) / 256, 256, 0, stream>>>(rs, out);
}